// GCN_3221225472201
// MI455X (gfx1250) — compile-verified
//
#include <hip/hip_runtime.h>

typedef __attribute__((ext_vector_type(2))) float v2f;
typedef __attribute__((ext_vector_type(8))) float v8f;

#define WAVES_PER_BLOCK 8
#define BLOCK_THREADS 256

// C[M,N] = A[M,K] @ B (+ bias), using V_WMMA_F32_16X16X4_F32.
//  transB=0: B is [K,N] row-major (C = A @ B)
//  transB=1: B is [N,K] row-major (C = A @ B^T)
// Requires M % 16 == 0, K % 4 == 0, K <= 256. N may be < 16*gridDim.y:
// the LDS B tile is zero-padded and stores are guarded.
__global__ __launch_bounds__(BLOCK_THREADS)
void gemm16_wmma_f32(const float* __restrict__ A, const float* __restrict__ B,
                     const float* __restrict__ bias, float* __restrict__ C,
                     int M, int K, int N, int transB, int hasBias)
{
    __shared__ float Bs[256 * 16];   // [k][n] tile for this block's 16 columns
    const int tid   = threadIdx.x;
    const int nbase = blockIdx.y * 16;

    // Cooperatively stage (and transpose if needed) the 16-wide B tile into LDS.
    for (int i = tid; i < K * 16; i += BLOCK_THREADS) {
        int k  = i >> 4;
        int n  = i & 15;
        int gn = nbase + n;
        float v = 0.0f;
        if (gn < N) v = transB ? B[(long)gn * K + k] : B[(long)k * N + gn];
        Bs[i] = v;
    }
    __syncthreads();

    const int  wave  = tid >> 5;          // wave32
    const int  lane  = tid & 31;
    const long mtile = (long)blockIdx.x * WAVES_PER_BLOCK + wave;
    if (mtile * 16 >= M) return;          // wave-uniform exit (EXEC all-1s at WMMA)

    const int mn   = lane & 15;           // A row in tile == D column in tile
    const int half = lane >> 4;

    const float*  arow  = A + (mtile * 16 + mn) * (long)K;
    const float2* arow2 = (const float2*)arow;

    v8f acc = {};
    for (int k = 0; k < K; k += 4) {
        const int kb = k + 2 * half;      // even -> contiguous float2 of A
        float2 a2 = arow2[kb >> 1];
        v2f av; av.x = a2.x; av.y = a2.y;                 // A[mn][kb], A[mn][kb+1]
        v2f bv; bv.x = Bs[(kb + 0) * 16 + mn];            // B[kb  ][mn]
                bv.y = Bs[(kb + 1) * 16 + mn];            // B[kb+1][mn]
        acc = __builtin_amdgcn_wmma_f32_16x16x4_f32(
                  /*neg_a=*/false, av, /*neg_b=*/false, bv,
                  /*c_mod=*/(short)0, acc, /*reuse_a=*/false, /*reuse_b=*/false);
    }

    const int gn = nbase + mn;
    if (gn < N) {
        const float bval = hasBias ? bias[gn] : 0.0f;
        #pragma unroll
        for (int r = 0; r < 8; ++r) {
            long row = mtile * 16 + r + 8 * half;         // D layout: M = r + 8*half
            C[row * (long)N + gn] = acc[r] + bval;
        }
    }
}

// One wave per edge: gather source row (float4 vector loads), scale by edge
// weight, scatter-add into destination row with global_atomic_add_f32.
__global__ __launch_bounds__(BLOCK_THREADS)
void spmm_scatter_f32(const int* __restrict__ rows, const int* __restrict__ cols,
                      const float* __restrict__ ew, const float* __restrict__ dense,
                      float* __restrict__ out, int nE, int F)
{
    const int warp = blockIdx.x * (BLOCK_THREADS >> 5) + (threadIdx.x >> 5);
    const int lane = threadIdx.x & 31;
    if (warp >= nE) return;
    const int   dst = rows[warp];
    const int   src = cols[warp];
    const float w   = ew[warp];
    const float4* s = (const float4*)(dense + (long)src * F);
    float*        d = out + (long)dst * F;
    for (int i = lane; i < (F >> 2); i += 32) {
        float4 v = s[i];
        atomicAdd(d + 4 * i + 0, v.x * w);
        atomicAdd(d + 4 * i + 1, v.y * w);
        atomicAdd(d + 4 * i + 2, v.z * w);
        atomicAdd(d + 4 * i + 3, v.w * w);
    }
}

__global__ void zero_f32(float* __restrict__ p, long n)
{
    long i      = (long)blockIdx.x * blockDim.x + threadIdx.x;
    long stride = (long)gridDim.x * blockDim.x;
    for (; i < n; i += stride) p[i] = 0.0f;
}

// p[r][f] = relu(p[r][f] + b[f]); F is a power of two (64 or 128).
__global__ void bias_relu_f32(float* __restrict__ p, const float* __restrict__ b,
                              long nrows, int F)
{
    long n      = nrows * (long)F;
    long i      = (long)blockIdx.x * blockDim.x + threadIdx.x;
    long stride = (long)gridDim.x * blockDim.x;
    for (; i < n; i += stride) {
        float v = p[i] + b[(int)(i & (long)(F - 1))];
        p[i] = v > 0.0f ? v : 0.0f;
    }
}

static inline int mtiles_blocks(int M) {
    return ((M / 16) + WAVES_PER_BLOCK - 1) / WAVES_PER_BLOCK;
}

extern "C" void kernel_launch(void* const* d_in, const int* in_sizes, int n_in,
                              void* d_out, int out_size, void* d_ws, size_t ws_size,
                              hipStream_t stream)
{
    const float* feat0 = (const float*)d_in[0];
    const float* feat1 = (const float*)d_in[1];
    const float* fc0_w = (const float*)d_in[2];
    const float* fc0_b = (const float*)d_in[3];
    const float* fc1_w = (const float*)d_in[4];
    const float* fc1_b = (const float*)d_in[5];
    const float* W1    = (const float*)d_in[6];
    const float* b1    = (const float*)d_in[7];
    const float* W2    = (const float*)d_in[8];
    const float* b2    = (const float*)d_in[9];
    const float* pw    = (const float*)d_in[10];   // [8,64]
    const float* pb    = (const float*)d_in[11];   // [8]
    const int*   erow  = (const int*)d_in[12];
    const int*   ecol  = (const int*)d_in[13];
    const float* ew    = (const float*)d_in[14];
    float*       out   = (float*)d_out;

    const int E       = in_sizes[12];
    const int N_NODES = 100000;
    const int NT      = 50000;

    // Workspace: 3 x [100000,128] f32 buffers (reused for layer 2).
    float* x   = (float*)d_ws;                    // projected features / h2
    float* h   = x + (long)N_NODES * 128;         // x@W1 / agg2
    float* agg = h + (long)N_NODES * 128;         // agg1 (-> relu'd x2)

    // Per-type projections: x = feat @ fc_w.T + fc_b  (transB=1)
    gemm16_wmma_f32<<<dim3(mtiles_blocks(NT), 8), BLOCK_THREADS, 0, stream>>>(
        feat0, fc0_w, fc0_b, x, NT, 128, 128, 1, 1);
    gemm16_wmma_f32<<<dim3(mtiles_blocks(NT), 8), BLOCK_THREADS, 0, stream>>>(
        feat1, fc1_w, fc1_b, x + (long)NT * 128, NT, 256, 128, 1, 1);

    // GCN layer 1: relu(spmm(adj, x @ W1) + b1)
    gemm16_wmma_f32<<<dim3(mtiles_blocks(N_NODES), 8), BLOCK_THREADS, 0, stream>>>(
        x, W1, nullptr, h, N_NODES, 128, 128, 0, 0);
    zero_f32<<<2048, 256, 0, stream>>>(agg, (long)N_NODES * 128);
    spmm_scatter_f32<<<(E + 7) / 8, BLOCK_THREADS, 0, stream>>>(
        erow, ecol, ew, h, agg, E, 128);
    bias_relu_f32<<<2048, 256, 0, stream>>>(agg, b1, N_NODES, 128);

    // GCN layer 2: relu(spmm(adj, x2 @ W2) + b2)  (reuse x as h2, h as agg2)
    gemm16_wmma_f32<<<dim3(mtiles_blocks(N_NODES), 4), BLOCK_THREADS, 0, stream>>>(
        agg, W2, nullptr, x, N_NODES, 128, 64, 0, 0);
    zero_f32<<<2048, 256, 0, stream>>>(h, (long)N_NODES * 64);
    spmm_scatter_f32<<<(E + 7) / 8, BLOCK_THREADS, 0, stream>>>(
        erow, ecol, ew, x, h, E, 64);
    bias_relu_f32<<<2048, 256, 0, stream>>>(h, b2, N_NODES, 64);

    // Predict head: out = x3 @ pred_w.T + pred_b (N=8, padded B tile)
    gemm16_wmma_f32<<<dim3(mtiles_blocks(N_NODES), 1), BLOCK_THREADS, 0, stream>>>(
        h, pw, pb, out, N_NODES, 64, 8, 1, 1);
}